// GNNModel_10299331576310
// MI455X (gfx1250) — compile-verified
//
#include <hip/hip_runtime.h>
#include <hip/hip_bf16.h>
#include <stdint.h>

// ---------------- problem constants ----------------
constexpr int KB    = 16;        // graphs
constexpr int KT    = 50;        // cell types
constexpr int KC    = 200;       // neurons per type
constexpr int KN    = KT * KC;   // 10000 nodes per graph
constexpr int KNT   = KB * KN;   // 160000 total nodes
constexpr int KH    = 8;         // heads
constexpr int KDEG  = 32;
constexpr int KE    = KNT * KDEG;    // 5,120,000 edges
constexpr int KETOT = KE + KNT;      // + self loops
constexpr int KPASS = 4;
constexpr float KEPS   = 1e-5f;
constexpr float KSLOPE = 0.2f;

typedef __attribute__((ext_vector_type(2)))  float    v2f;
typedef __attribute__((ext_vector_type(8)))  float    v8f;
typedef __attribute__((ext_vector_type(4)))  uint32_t v4u;
typedef __attribute__((ext_vector_type(4)))  int      v4i;
typedef __attribute__((ext_vector_type(8)))  int      v8i;

// Toolchain discriminator: therock-10.0 headers ship the TDM header and the
// 6-arg tensor builtin; ROCm 7.2 has the 5-arg form.
#if __has_include(<hip/amd_detail/amd_gfx1250_TDM.h>)
#define TDM_6ARG 1
#else
#define TDM_6ARG 0
#endif

// ---------------- threefry2x32 (20 rounds, JAX-compatible core) ----------------
__host__ __device__ inline void tf2x32(uint32_t k0, uint32_t k1,
                                       uint32_t c0, uint32_t c1,
                                       uint32_t& o0, uint32_t& o1) {
  uint32_t ks2 = 0x1BD11BDAu ^ k0 ^ k1;
  uint32_t x0 = c0 + k0, x1 = c1 + k1;
#define TFR(r) { x0 += x1; x1 = (x1 << (r)) | (x1 >> (32 - (r))); x1 ^= x0; }
  TFR(13) TFR(15) TFR(26) TFR(6)  x0 += k1;  x1 += ks2 + 1u;
  TFR(17) TFR(29) TFR(16) TFR(24) x0 += ks2; x1 += k0 + 2u;
  TFR(13) TFR(15) TFR(26) TFR(6)  x0 += k0;  x1 += k1 + 3u;
  TFR(17) TFR(29) TFR(16) TFR(24) x0 += k1;  x1 += ks2 + 4u;
  TFR(13) TFR(15) TFR(26) TFR(6)  x0 += ks2; x1 += k0 + 5u;
#undef TFR
  o0 = x0; o1 = x1;
}

__device__ inline float u01f(uint32_t bits) {
  return __uint_as_float(0x3F800000u | (bits >> 9)) - 1.0f;
}

__device__ inline float gumbel_at(uint32_t kx, uint32_t ky, uint32_t idx) {
  uint32_t r0, r1;
  tf2x32(kx, ky, idx, 0x9E3779B9u, r0, r1);
  float u = fmaxf(u01f(r0), 1e-10f);
  return -logf(-logf(u));
}

// ---------------- monotone float <-> uint key (for atomicMax on floats) -------
__device__ inline unsigned fkey(float f) {
  unsigned b = __float_as_uint(f);
  return b ^ ((unsigned)((int)b >> 31) | 0x80000000u);
}
__device__ inline float fdec(unsigned k) {
  unsigned b = (k & 0x80000000u) ? (k ^ 0x80000000u) : ~k;
  return __uint_as_float(b);
}
constexpr unsigned KEY_NEGINF = 0x007FFFFFu; // fkey(-inf)

// ---------------- Tensor Data Mover: 2D tile (width=KC f32, `rows` rows) ------
// D# per CDNA5 ISA ch.8: group0 = {count/type/addrs}, group1 = {dims/strides}.
__device__ inline void tdm_load_2d(uint32_t lds_addr, const void* gptr, int rows) {
  uint64_t ga = (uint64_t)(uintptr_t)gptr;
  v4u g0;
  g0.x = 1u;                                              // count=1, user D#
  g0.y = lds_addr;                                        // LDS dest (bytes)
  g0.z = (uint32_t)(ga & 0xFFFFFFFFu);                    // global_addr[31:0]
  g0.w = (uint32_t)((ga >> 32) & 0x01FFFFFFu) | (2u << 30); // addr[56:32], type=2
  v8i g1;
  g1[0] = (int)(2u << 16);                                // data_size = 4 bytes
  g1[1] = (int)(((uint32_t)KC & 0xFFFFu) << 16);          // tensor_dim0 = 200
  g1[2] = (int)(((uint32_t)rows & 0xFFFFu) << 16);        // tensor_dim1 = rows
  g1[3] = (int)(((uint32_t)KC & 0xFFFFu) << 16);          // tile_dim0 = 200
  g1[4] = (int)((uint32_t)rows & 0xFFFFu);                // tile_dim1 = rows
  g1[5] = KC;                                             // dim0 stride = 200
  g1[6] = 0;
  g1[7] = 0;
  v4i z4 = {0, 0, 0, 0};
#if TDM_6ARG
  v8i z8 = {0, 0, 0, 0, 0, 0, 0, 0};
  __builtin_amdgcn_tensor_load_to_lds(g0, g1, z4, z4, z8, 0);
#else
  __builtin_amdgcn_tensor_load_to_lds(g0, g1, z4, z4, 0);
#endif
}

// =============================================================================
// Kernel A: permutation layer.  One block per (t,b) matrix; 4 waves; each wave
// owns a 16x200 LDS tile.  Per 16-row block:
//   TDM DMA-loads the weight slice into LDS  ->  s_wait_tensorcnt 0
//   sweep1: s = w + gumbel in place, rowmax (2 lanes/row)
//   sweep2: exp in place
//   WMMA f32 16x16x4 chain: D[:,0]=rowsum, D[:,1]=dot with x (B col0=1, col1=x)
// =============================================================================
__global__ __launch_bounds__(128) void perm_kernel(
    const float* __restrict__ x, const float* __restrict__ pw,
    float* __restrict__ y, uint32_t kgx, uint32_t kgy) {
  __shared__ float sx[KC];
  __shared__ float Wt[4][16][KC];    // per-wave weight/score tile (TDM dest)
  __shared__ float rbuf[4][2][16];   // [wave][sum|dot][row]

  const int tb = blockIdx.x;           // t*KB + b
  const int t  = tb / KB, b = tb % KB;
  const float* __restrict__ pwb = pw + (size_t)tb * KC * KC;
  const int gx = b * KN + t * KC;      // node block base for this (t,b)

  for (int j = threadIdx.x; j < KC; j += blockDim.x) sx[j] = x[gx + j];
  __syncthreads();

  const int wid  = threadIdx.x >> 5;
  const int lane = threadIdx.x & 31;
  const int m    = lane & 15;          // row within 16-row block / N column of D
  const int half = lane >> 4;          // 0: K=0,1  1: K=2,3 (A/B VGPR layout)
  const uint32_t wt_lds = (uint32_t)(uintptr_t)&Wt[wid][0][0];
  const float is0 = (m == 0) ? 1.0f : 0.0f;   // B col 0 selector (ones)
  const float is1 = (m == 1) ? 1.0f : 0.0f;   // B col 1 selector (x)

  for (int rb = wid; rb < (KC + 15) / 16; rb += 4) {   // uniform per wave
    const int row   = rb * 16 + m;
    const bool valid = row < KC;
    const int rows  = min(16, KC - rb * 16);

    // all prior LDS reads of this wave's tile must retire before DMA overwrites
    asm volatile("s_wait_dscnt 0" ::: "memory");
    tdm_load_2d(wt_lds, pwb + (size_t)rb * 16 * KC, rows);
    __builtin_amdgcn_s_wait_tensorcnt(0);

    // --- sweep 1: s = w + gumbel in place, track row max (2 lanes per row)
    float mx = -3.4e38f;
    for (int j = half; j < KC; j += 2) {
      float w = Wt[wid][m][j];
      float g = gumbel_at(kgx, kgy, (uint32_t)(tb * KC * KC + row * KC + j));
      float s = w + g;                  // TAU == 1
      Wt[wid][m][j] = s;
      mx = fmaxf(mx, s);
    }
    mx = fmaxf(mx, __shfl_xor(mx, 16, 32));

    // --- sweep 2: exponentiate in place (pad rows -> 0)
    for (int j = half; j < KC; j += 2) {
      float e = valid ? expf(Wt[wid][m][j] - mx) : 0.0f;
      Wt[wid][m][j] = e;
    }
    __threadfence_block();

    // --- WMMA accumulation: D[:,0] = rowsum(E), D[:,1] = E . x  (branchless B)
    v8f acc = {};
    for (int c = 0; c < KC / 4; ++c) {
      const int k0 = 4 * c + (half ? 2 : 0);
      v2f a = *(const v2f*)&Wt[wid][m][k0];   // 8B aligned: even stride + even k0
      v2f bv;
      bv.x = fmaf(is1, sx[k0],     is0);      // lane0:(1,1) lane1:(x,x) else 0
      bv.y = fmaf(is1, sx[k0 + 1], is0);
      acc = __builtin_amdgcn_wmma_f32_16x16x4_f32(
          false, a, false, bv, (short)0, acc, false, false);
    }

    // lanes 0/16 hold D[:,0] (rowsum), lanes 1/17 hold D[:,1] (dot)
    if (m < 2) {
      const int rbase = half ? 8 : 0;
#pragma unroll
      for (int r = 0; r < 8; ++r) rbuf[wid][m][rbase + r] = acc[r];
    }
    __threadfence_block();
    if (lane < 16) {
      const int row2 = rb * 16 + lane;
      if (row2 < KC) y[gx + row2] = rbuf[wid][1][lane] / rbuf[wid][0][lane];
    }
    __threadfence_block();
  }
}

// =============================================================================
// Kernel B: nonzero count / sum / sumsq reduction over y
// =============================================================================
__global__ __launch_bounds__(256) void stats_kernel(const float* __restrict__ y,
                                                    float* __restrict__ stats) {
  float c = 0.f, s = 0.f, ss = 0.f;
  for (int i = blockIdx.x * blockDim.x + threadIdx.x; i < KNT;
       i += gridDim.x * blockDim.x) {
    float v = y[i];
    if (v != 0.0f) { c += 1.0f; s += v; ss += v * v; }
  }
#pragma unroll
  for (int o = 16; o > 0; o >>= 1) {
    c  += __shfl_xor(c,  o, 32);
    s  += __shfl_xor(s,  o, 32);
    ss += __shfl_xor(ss, o, 32);
  }
  if ((threadIdx.x & 31) == 0) {
    atomicAdd(&stats[0], c);
    atomicAdd(&stats[1], s);
    atomicAdd(&stats[2], ss);
  }
}

// =============================================================================
// Kernel C: inject (std/100)*N(0,1) noise into zero entries (Box-Muller)
// =============================================================================
__global__ __launch_bounds__(256) void noise_kernel(
    const float* __restrict__ y, float* __restrict__ xc,
    const float* __restrict__ stats, uint32_t knx, uint32_t kny) {
  int i = blockIdx.x * blockDim.x + threadIdx.x;
  if (i >= KNT) return;
  float v = y[i];
  if (v != 0.0f) { xc[i] = v; return; }
  float n    = stats[0];
  float mean = stats[1] / n;
  float var  = (stats[2] - n * mean * mean) / (n - 1.0f);
  float sd   = sqrtf(fmaxf(var, 0.0f));
  uint32_t r0, r1;
  tf2x32(knx, kny, (uint32_t)i, 0x243F6A88u, r0, r1);
  float u1 = fmaxf(u01f(r0), 1e-7f);
  float u2 = u01f(r1);
  float z = sqrtf(-2.0f * logf(u1)) * cosf(6.28318530718f * u2);
  xc[i] = (sd * 0.01f) * z;
}

// =============================================================================
// Kernel I: initialize workspace accumulators
// =============================================================================
__global__ __launch_bounds__(256) void init_kernel(
    unsigned* __restrict__ emaxK, float* __restrict__ denom,
    float* __restrict__ num, float* __restrict__ stats,
    unsigned* __restrict__ poolK) {
  int i = blockIdx.x * blockDim.x + threadIdx.x;
  if (i < KNT * KH) { emaxK[i] = KEY_NEGINF; denom[i] = 0.0f; num[i] = 0.0f; }
  if (i < 3)  stats[i] = 0.0f;
  if (i < KB) poolK[i] = 0u;   // below fkey(any finite) we will produce
}

// =============================================================================
// Kernel D: edge sweep 1 — per-dst per-head attention max (monotone-key atomics)
// e[h] = leakyrelu(x[src]*c1[h] + x[dst]*c2[h]),  c1=w*att_src, c2=w*att_dst
// =============================================================================
__global__ __launch_bounds__(256) void gat_max_kernel(
    const int* __restrict__ ei, const float* __restrict__ xc,
    unsigned* __restrict__ emaxK, const float* __restrict__ gw,
    const float* __restrict__ asr, const float* __restrict__ adt) {
  int idx = blockIdx.x * blockDim.x + threadIdx.x;
  if (idx >= KETOT) return;
  if (idx + 8192 < KE) __builtin_prefetch(&ei[idx + 8192], 0, 1);
  int s, d;
  if (idx < KE) { s = ei[idx]; d = ei[KE + idx]; }
  else          { s = d = idx - KE; }
  float xs = xc[s], xd = xc[d];
#pragma unroll
  for (int h = 0; h < KH; ++h) {
    float e = xs * (gw[h] * asr[h]) + xd * (gw[h] * adt[h]);
    e = (e > 0.0f) ? e : KSLOPE * e;
    atomicMax(&emaxK[d * KH + h], fkey(e));
  }
}

// =============================================================================
// Kernel E: edge sweep 2 — ee = exp(e - emax[dst]); denom += ee; num += ee*xs*w
// =============================================================================
__global__ __launch_bounds__(256) void gat_sum_kernel(
    const int* __restrict__ ei, const float* __restrict__ xc,
    const unsigned* __restrict__ emaxK, float* __restrict__ denom,
    float* __restrict__ num, const float* __restrict__ gw,
    const float* __restrict__ asr, const float* __restrict__ adt) {
  int idx = blockIdx.x * blockDim.x + threadIdx.x;
  if (idx >= KETOT) return;
  if (idx + 8192 < KE) __builtin_prefetch(&ei[idx + 8192], 0, 1);
  int s, d;
  if (idx < KE) { s = ei[idx]; d = ei[KE + idx]; }
  else          { s = d = idx - KE; }
  float xs = xc[s], xd = xc[d];
#pragma unroll
  for (int h = 0; h < KH; ++h) {
    float e = xs * (gw[h] * asr[h]) + xd * (gw[h] * adt[h]);
    e = (e > 0.0f) ? e : KSLOPE * e;
    float ee = expf(e - fdec(emaxK[d * KH + h]));
    atomicAdd(&denom[d * KH + h], ee);
    atomicAdd(&num[d * KH + h], ee * xs * gw[h]);
  }
}

// =============================================================================
// Kernel F: node update = mean_h(num/denom)+bias -> BN -> silu -> +res; also
// resets the per-head accumulators for the next pass.
// =============================================================================
__global__ __launch_bounds__(256) void node_update_kernel(
    float* __restrict__ xc, const float* __restrict__ yres,
    unsigned* __restrict__ emaxK, float* __restrict__ denom,
    float* __restrict__ num, const float* __restrict__ bias,
    const float* __restrict__ bng, const float* __restrict__ bnb,
    const float* __restrict__ bnm, const float* __restrict__ bnv,
    float persist_i) {
  int n = blockIdx.x * blockDim.x + threadIdx.x;
  if (n >= KNT) return;
  float acc = 0.0f;
#pragma unroll
  for (int h = 0; h < KH; ++h) {
    int k = n * KH + h;
    acc += num[k] / denom[k];
    emaxK[k] = KEY_NEGINF; denom[k] = 0.0f; num[k] = 0.0f;  // reset for next pass
  }
  float v = acc * (1.0f / KH) + bias[0];
  v = (v - bnm[0]) * rsqrtf(bnv[0] + KEPS) * bng[0] + bnb[0];
  v = v / (1.0f + expf(-v));                 // silu
  v += yres[n] * persist_i;
  xc[n] = v;
}

// =============================================================================
// Kernel G: mask + per-graph max pool (monotone-key atomics)
// =============================================================================
__global__ __launch_bounds__(256) void pool_kernel(
    const float* __restrict__ xc, const float* __restrict__ dec,
    unsigned* __restrict__ poolK) {
  int n = blockIdx.x * blockDim.x + threadIdx.x;
  if (n >= KNT) return;
  float v = xc[n] * dec[n % KN];
  atomicMax(&poolK[n / KN], fkey(v));
}

// =============================================================================
// Kernel H: relu(pool @ lin_w + lin_b)
// =============================================================================
__global__ __launch_bounds__(32) void out_kernel(
    const unsigned* __restrict__ poolK, const float* __restrict__ lw,
    const float* __restrict__ lb, float* __restrict__ out) {
  int b = threadIdx.x;
  if (b >= KB) return;
  float v = fdec(poolK[b]) * lw[0] + lb[0];
  out[b] = v > 0.0f ? v : 0.0f;
}

// =============================================================================
extern "C" void kernel_launch(void* const* d_in, const int* in_sizes, int n_in,
                              void* d_out, int out_size, void* d_ws, size_t ws_size,
                              hipStream_t stream) {
  const float* x    = (const float*)d_in[0];
  const int*   ei   = (const int*)d_in[1];
  const float* pw   = (const float*)d_in[3];
  const float* dec  = (const float*)d_in[4];
  const float* gw   = (const float*)d_in[5];
  const float* asr  = (const float*)d_in[6];
  const float* adt  = (const float*)d_in[7];
  const float* gb   = (const float*)d_in[8];
  const float* bng  = (const float*)d_in[9];
  const float* bnb  = (const float*)d_in[10];
  const float* bnm  = (const float*)d_in[11];
  const float* bnv  = (const float*)d_in[12];
  const float* lw   = (const float*)d_in[13];
  const float* lb   = (const float*)d_in[14];
  float* out = (float*)d_out;

  // workspace layout (floats)
  float*    ws    = (float*)d_ws;
  float*    y     = ws;                         // KNT   (x_res)
  float*    xc    = ws + KNT;                   // KNT
  unsigned* emaxK = (unsigned*)(ws + 2 * KNT);  // KNT*KH
  float*    denom = ws + 2 * KNT + KNT * KH;    // KNT*KH
  float*    num   = ws + 2 * KNT + 2 * KNT * KH;// KNT*KH
  float*    stats = ws + 2 * KNT + 3 * KNT * KH;// 3
  unsigned* poolK = (unsigned*)(stats + 4);     // KB

  // JAX-style keys: rk = key(42); fold_in(rk,0) -> gumbel; fold_in(rk,1) -> noise
  uint32_t kg0, kg1, kn0, kn1;
  tf2x32(0u, 42u, 0u, 0u, kg0, kg1);
  tf2x32(0u, 42u, 0u, 1u, kn0, kn1);

  const int nodeBlocks = (KNT + 255) / 256;
  const int initBlocks = (KNT * KH + 255) / 256;
  const int edgeBlocks = (KETOT + 255) / 256;

  init_kernel<<<initBlocks, 256, 0, stream>>>(emaxK, denom, num, stats, poolK);
  perm_kernel<<<KT * KB, 128, 0, stream>>>(x, pw, y, kg0, kg1);
  stats_kernel<<<nodeBlocks, 256, 0, stream>>>(y, stats);
  noise_kernel<<<nodeBlocks, 256, 0, stream>>>(y, xc, stats, kn0, kn1);

  for (int p = 0; p < KPASS; ++p) {
    gat_max_kernel<<<edgeBlocks, 256, 0, stream>>>(ei, xc, emaxK, gw, asr, adt);
    gat_sum_kernel<<<edgeBlocks, 256, 0, stream>>>(ei, xc, emaxK, denom, num,
                                                   gw, asr, adt);
    node_update_kernel<<<nodeBlocks, 256, 0, stream>>>(
        xc, y, emaxK, denom, num, gb, bng, bnb, bnm, bnv, 1.0f /* PERSIST^p */);
  }

  pool_kernel<<<nodeBlocks, 256, 0, stream>>>(xc, dec, poolK);
  out_kernel<<<1, 32, 0, stream>>>(poolK, lw, lb, out);
}